// DetectionTargetLayer_5317169512507
// MI455X (gfx1250) — compile-verified
//
#include <hip/hip_runtime.h>
#include <stdint.h>

#define N_PROP      2000
#define MAX_GT      50
#define IMG_H       1024
#define IMG_W       1024
#define TRAIN_ROIS  200
#define POS_CAP     66
#define NEG_CAP     134
#define MASK_H      28
#define MASK_W      28
#define SCORE_PITCH 2048

typedef float v2f __attribute__((ext_vector_type(2)));
typedef float v8f __attribute__((ext_vector_type(8)));

// ---------------- JAX threefry2x32 (20 rounds) ----------------
__device__ __forceinline__ void tf_block(uint32_t k0, uint32_t k1,
                                         uint32_t x0, uint32_t x1,
                                         uint32_t& o0, uint32_t& o1) {
  uint32_t ks2 = k0 ^ k1 ^ 0x1BD11BDAu;
#define ROTL(x, r) (((x) << (r)) | ((x) >> (32 - (r))))
#define R4(a, b, c, d)                                 \
  x0 += x1; x1 = ROTL(x1, a); x1 ^= x0;                \
  x0 += x1; x1 = ROTL(x1, b); x1 ^= x0;                \
  x0 += x1; x1 = ROTL(x1, c); x1 ^= x0;                \
  x0 += x1; x1 = ROTL(x1, d); x1 ^= x0;
  x0 += k0; x1 += k1;
  R4(13, 15, 26, 6);  x0 += k1;  x1 += ks2 + 1u;
  R4(17, 29, 16, 24); x0 += ks2; x1 += k0 + 2u;
  R4(13, 15, 26, 6);  x0 += k0;  x1 += k1 + 3u;
  R4(17, 29, 16, 24); x0 += k1;  x1 += ks2 + 4u;
  R4(13, 15, 26, 6);  x0 += ks2; x1 += k0 + 5u;
  o0 = x0; o1 = x1;
#undef R4
#undef ROTL
}

__device__ __forceinline__ float bits_to_uniform(uint32_t bits) {
  uint32_t u = (bits >> 9) | 0x3f800000u;
  return fmaxf(__uint_as_float(u) - 1.0f, 0.0f);
}

__device__ __forceinline__ float iou_fn(float ay1, float ax1, float ay2, float ax2,
                                        float by1, float bx1, float by2, float bx2) {
  float iy = fmaxf(fminf(ay2, by2) - fmaxf(ay1, by1), 0.0f);
  float ix = fmaxf(fminf(ax2, bx2) - fmaxf(ax1, bx1), 0.0f);
  float inter = iy * ix;
  float uni = (ay2 - ay1) * (ax2 - ax1) + (by2 - by1) * (bx2 - bx1) - inter;
  return uni > 0.0f ? inter / uni : 0.0f;
}

// ---------------- Stage 1: per-proposal masks + random scores ----------------
__global__ void k_score(const float* __restrict__ proposals,
                        const int* __restrict__ gt_cls,
                        const float* __restrict__ gt_boxes,
                        float* __restrict__ pos_score,
                        float* __restrict__ neg_score) {
  const int b = blockIdx.y;
  __shared__ float sB[MAX_GT][4];
  __shared__ int   sC[MAX_GT];
  for (int t = threadIdx.x; t < MAX_GT; t += blockDim.x) {
    const float* gb = gt_boxes + ((size_t)b * MAX_GT + t) * 4;
    sB[t][0] = gb[0]; sB[t][1] = gb[1]; sB[t][2] = gb[2]; sB[t][3] = gb[3];
    sC[t] = gt_cls[b * MAX_GT + t];
  }
  __syncthreads();
  const int i = blockIdx.x * blockDim.x + threadIdx.x;
  if (i >= N_PROP) return;

  const float* pr = proposals + ((size_t)b * N_PROP + i) * 4;
  float py1 = pr[0], px1 = pr[1], py2 = pr[2], px2 = pr[3];
  bool validP = (fabsf(py1) + fabsf(px1) + fabsf(py2) + fabsf(px2)) > 0.0f;

  float roiMax = -1.0f, crowdMax = -1.0f;
  for (int g = 0; g < MAX_GT; ++g) {
    float gy1 = sB[g][0], gx1 = sB[g][1], gy2 = sB[g][2], gx2 = sB[g][3];
    bool validG = (fabsf(gy1) + fabsf(gx1) + fabsf(gy2) + fabsf(gx2)) > 0.0f;
    int c = sC[g];
    float ov = iou_fn(py1, px1, py2, px2, gy1, gx1, gy2, gx2);
    roiMax   = fmaxf(roiMax,   (c > 0 && validG) ? ov : -1.0f);
    crowdMax = fmaxf(crowdMax, (c < 0 && validG) ? ov : -1.0f);
  }
  bool posm = validP && (roiMax >= 0.5f);
  bool negm = validP && (roiMax < 0.5f) && (crowdMax < 1e-3f);

  // key chain: key(42) -> split(B)[b] -> split -> (kp, kn); partitionable draws.
  uint32_t kb0, kb1, kp0, kp1, kn0, kn1, o0, o1;
  tf_block(0u, 42u, 0u, (uint32_t)b, kb0, kb1);
  tf_block(kb0, kb1, 0u, 0u, kp0, kp1);
  tf_block(kb0, kb1, 0u, 1u, kn0, kn1);
  tf_block(kp0, kp1, 0u, (uint32_t)i, o0, o1);
  float up = bits_to_uniform(o0 ^ o1);
  tf_block(kn0, kn1, 0u, (uint32_t)i, o0, o1);
  float un = bits_to_uniform(o0 ^ o1);

  pos_score[b * SCORE_PITCH + i] = posm ? up : -1.0f;
  neg_score[b * SCORE_PITCH + i] = negm ? un : -1.0f;
}

// ---------------- Stage 2: stable top-k selection via ranks ----------------
__global__ void k_select(const float* __restrict__ pos_score,
                         const float* __restrict__ neg_score,
                         int* __restrict__ pos_idx, int* __restrict__ neg_idx,
                         int* __restrict__ pn) {
  const int b = blockIdx.x;
  const float* ps = pos_score + b * SCORE_PITCH;
  const float* ns = neg_score + b * SCORE_PITCH;
  __shared__ int red[256];
  __shared__ int shPos, shNeg;

  int cp = 0, cn = 0;
  for (int i = threadIdx.x; i < N_PROP; i += 256) {
    cp += (ps[i] >= 0.0f);
    cn += (ns[i] >= 0.0f);
  }
  red[threadIdx.x] = cp; __syncthreads();
  for (int s = 128; s > 0; s >>= 1) {
    if (threadIdx.x < s) red[threadIdx.x] += red[threadIdx.x + s];
    __syncthreads();
  }
  if (threadIdx.x == 0) shPos = red[0];
  __syncthreads();
  red[threadIdx.x] = cn; __syncthreads();
  for (int s = 128; s > 0; s >>= 1) {
    if (threadIdx.x < s) red[threadIdx.x] += red[threadIdx.x + s];
    __syncthreads();
  }
  if (threadIdx.x == 0) shNeg = red[0];
  __syncthreads();

  // rank = #elements strictly better (desc score, asc index tiebreak) == top_k order
  for (int i = threadIdx.x; i < N_PROP; i += 256) {
    float s = ps[i];
    if (s >= 0.0f) {
      int r = 0;
      for (int j = 0; j < N_PROP; ++j) {
        float t = ps[j];
        r += (t > s) || ((t == s) && (j < i));
      }
      if (r < POS_CAP) pos_idx[b * POS_CAP + r] = i;
    }
    float s2 = ns[i];
    if (s2 >= 0.0f) {
      int r = 0;
      for (int j = 0; j < N_PROP; ++j) {
        float t = ns[j];
        r += (t > s2) || ((t == s2) && (j < i));
      }
      if (r < NEG_CAP) neg_idx[b * NEG_CAP + r] = i;
    }
  }
  if (threadIdx.x == 0) {
    int p = min(shPos, POS_CAP);
    const float inv_ratio = (float)(1.0 / 0.33);   // float32(3.0303030...)
    int needed = (int)(inv_ratio * (float)p) - p;  // truncation == astype(int32)
    int n = min(min(shNeg, needed), NEG_CAP);
    pn[2 * b] = p;
    pn[2 * b + 1] = n;
  }
}

// ---------------- Stage 3: pack rois/class/deltas; stash per-positive info ---
__global__ void k_pack(const float* __restrict__ proposals,
                       const int* __restrict__ gt_cls,
                       const float* __restrict__ gt_boxes,
                       const int* __restrict__ pos_idx,
                       const int* __restrict__ neg_idx,
                       const int* __restrict__ pn,
                       int* __restrict__ assign_ws,
                       float* __restrict__ roibox_ws,
                       float* __restrict__ out, int B) {
  const int b = blockIdx.x;
  float* rois = out + (size_t)b * TRAIN_ROIS * 4;
  float* cls  = out + (size_t)B * TRAIN_ROIS * 4 + (size_t)b * TRAIN_ROIS;
  float* dlt  = out + (size_t)B * TRAIN_ROIS * 5 + (size_t)b * TRAIN_ROIS * 4;

  for (int t = threadIdx.x; t < TRAIN_ROIS * 4; t += 256) { rois[t] = 0.0f; dlt[t] = 0.0f; }
  for (int t = threadIdx.x; t < TRAIN_ROIS; t += 256) cls[t] = 0.0f;
  __syncthreads();

  const int p = pn[2 * b], n = pn[2 * b + 1];
  const int t = threadIdx.x;
  if (t < POS_CAP) {
    if (t < p) {
      int idx = pos_idx[b * POS_CAP + t];
      const float* pr = proposals + ((size_t)b * N_PROP + idx) * 4;
      float y1 = pr[0], x1 = pr[1], y2 = pr[2], x2 = pr[3];
      // argmax over overlaps row (crowd/invalid -> -1); first-max like jnp.argmax
      float best = -2.0f; int bg = 0;
      for (int g = 0; g < MAX_GT; ++g) {
        const float* gb = gt_boxes + ((size_t)b * MAX_GT + g) * 4;
        float gy1 = gb[0], gx1 = gb[1], gy2 = gb[2], gx2 = gb[3];
        bool validG = (fabsf(gy1) + fabsf(gx1) + fabsf(gy2) + fabsf(gx2)) > 0.0f;
        int c = gt_cls[b * MAX_GT + g];
        float ov = (c > 0 && validG)
                 ? iou_fn(y1, x1, y2, x2, gy1, gx1, gy2, gx2) : -1.0f;
        if (ov > best) { best = ov; bg = g; }
      }
      const float* gb = gt_boxes + ((size_t)b * MAX_GT + bg) * 4;
      float gy1 = gb[0], gx1 = gb[1], gy2 = gb[2], gx2 = gb[3];
      float h = y2 - y1, w = x2 - x1;
      float cy = y1 + 0.5f * h, cx = x1 + 0.5f * w;
      float gh = gy2 - gy1, gw = gx2 - gx1;
      float gcy = gy1 + 0.5f * gh, gcx = gx1 + 0.5f * gw;
      rois[t * 4 + 0] = y1; rois[t * 4 + 1] = x1; rois[t * 4 + 2] = y2; rois[t * 4 + 3] = x2;
      cls[t] = (float)gt_cls[b * MAX_GT + bg];
      dlt[t * 4 + 0] = ((gcy - cy) / h) / 0.1f;
      dlt[t * 4 + 1] = ((gcx - cx) / w) / 0.1f;
      dlt[t * 4 + 2] = logf(gh / h) / 0.2f;
      dlt[t * 4 + 3] = logf(gw / w) / 0.2f;
      assign_ws[b * POS_CAP + t] = bg;
      float* rb = roibox_ws + ((size_t)b * POS_CAP + t) * 4;
      rb[0] = y1; rb[1] = x1; rb[2] = y2; rb[3] = x2;
    }
  } else if (t < POS_CAP + NEG_CAP) {
    int j = t - POS_CAP;
    if (j < n) {
      int idx = neg_idx[b * NEG_CAP + j];
      const float* pr = proposals + ((size_t)b * N_PROP + idx) * 4;
      int slot = p + j;
      rois[slot * 4 + 0] = pr[0]; rois[slot * 4 + 1] = pr[1];
      rois[slot * 4 + 2] = pr[2]; rois[slot * 4 + 3] = pr[3];
    }
  }
}

// ---------------- Stage 4: WMMA crop_and_resize ------------------------------
// O(32x32) = Wy(32x64) * S(64x64) * WxT(64x32), S = gathered 56x56 mask window,
// run on V_WMMA_F32_16X16X4_F32, one wave per ROI slot.
__global__ void __launch_bounds__(32)
k_masks(const float* __restrict__ gt_masks, const int* __restrict__ pn,
        const int* __restrict__ assign_ws, const float* __restrict__ roibox_ws,
        float* __restrict__ out_masks) {
  const int b = blockIdx.x / TRAIN_ROIS;
  const int k = blockIdx.x % TRAIN_ROIS;
  float* out = out_masks + (size_t)(b * TRAIN_ROIS + k) * (MASK_H * MASK_W);
  const int tid = threadIdx.x;
  const int p = pn[2 * b];
  if (k >= p) {
    for (int t = tid; t < MASK_H * MASK_W; t += 32) out[t] = 0.0f;
    return;
  }

  __shared__ float S[64 * 64];
  __shared__ float Wy[32 * 64];
  __shared__ float Xw[64 * 32];
  __shared__ float Tm[32 * 64];
  __shared__ int yi0[MASK_H], yi1[MASK_H], xi0[MASK_W], xi1[MASK_W];
  __shared__ int vy[MASK_H], vx[MASK_W];

  for (int t = tid; t < 64 * 64; t += 32) S[t] = 0.0f;
  for (int t = tid; t < 32 * 64; t += 32) Wy[t] = 0.0f;
  for (int t = tid; t < 64 * 32; t += 32) Xw[t] = 0.0f;
  __syncthreads();

  const int g = assign_ws[b * POS_CAP + k];
  const float* rb = roibox_ws + ((size_t)b * POS_CAP + k) * 4;
  const float y1 = rb[0], x1 = rb[1], y2 = rb[2], x2 = rb[3];

  if (tid < MASK_H) {
    float ys = y1 * (float)(IMG_H - 1) +
               (float)tid * ((y2 - y1) * (float)(IMG_H - 1) / (float)(MASK_H - 1));
    float y0 = floorf(ys);
    float wyv = ys - y0;
    yi0[tid] = (int)fminf(fmaxf(y0, 0.0f), (float)(IMG_H - 1));
    yi1[tid] = (int)fminf(fmaxf(y0 + 1.0f, 0.0f), (float)(IMG_H - 1));
    vy[tid] = (ys >= 0.0f) && (ys <= (float)(IMG_H - 1));
    Wy[tid * 64 + 2 * tid]     = 1.0f - wyv;
    Wy[tid * 64 + 2 * tid + 1] = wyv;

    float xs = x1 * (float)(IMG_W - 1) +
               (float)tid * ((x2 - x1) * (float)(IMG_W - 1) / (float)(MASK_W - 1));
    float x0 = floorf(xs);
    float wxv = xs - x0;
    xi0[tid] = (int)fminf(fmaxf(x0, 0.0f), (float)(IMG_W - 1));
    xi1[tid] = (int)fminf(fmaxf(x0 + 1.0f, 0.0f), (float)(IMG_W - 1));
    vx[tid] = (xs >= 0.0f) && (xs <= (float)(IMG_W - 1));
    Xw[(2 * tid) * 32 + tid]     = 1.0f - wxv;
    Xw[(2 * tid + 1) * 32 + tid] = wxv;
  }
  __syncthreads();

  // gather compacted 56x56 source window; gt_masks layout [B, H, W, G]
  const size_t base = (size_t)b * ((size_t)IMG_H * IMG_W * MAX_GT) + (size_t)g;
  for (int e = tid; e < 56 * 56; e += 32) {
    int r = e / 56, c = e % 56;
    int y = (r & 1) ? yi1[r >> 1] : yi0[r >> 1];
    int x = (c & 1) ? xi1[c >> 1] : xi0[c >> 1];
    S[r * 64 + c] = gt_masks[base + ((size_t)y * IMG_W + x) * MAX_GT];
  }
  __syncthreads();

  const int lr = tid & 15;       // lane within half
  const int lh = tid >> 4;       // half select
  // A 16x4 frag: lane<16: K k0,k0+1 ; lane>=16: K k0+2,k0+3 (ISA 7.12.2)
  auto loadA = [&](const float* M, int LD, int row0, int k0) {
    v2f a;
    int rr = row0 + lr, kk = k0 + (lh << 1);
    a.x = M[rr * LD + kk]; a.y = M[rr * LD + kk + 1];
    return a;
  };
  // B 4x16 frag: rows striped across lanes within a VGPR (C/D pattern)
  auto loadB = [&](const float* M, int LD, int k0, int col0) {
    v2f bb;
    int cc = col0 + lr, kk = k0 + (lh << 1);
    bb.x = M[kk * LD + cc]; bb.y = M[(kk + 1) * LD + cc];
    return bb;
  };

  // GEMM1: Tm(32x64) = Wy(32x64) * S(64x64)
  for (int mt = 0; mt < 2; ++mt) {
    for (int nt = 0; nt < 4; ++nt) {
      v8f acc = {};
      for (int ks = 0; ks < 16; ++ks) {
        v2f a  = loadA(Wy, 64, mt * 16, ks * 4);
        v2f bb = loadB(S, 64, ks * 4, nt * 16);
        acc = __builtin_amdgcn_wmma_f32_16x16x4_f32(false, a, false, bb,
                                                    (short)0, acc, false, false);
      }
      for (int q = 0; q < 8; ++q) {
        int row = mt * 16 + q + (lh << 3);
        int col = nt * 16 + lr;
        Tm[row * 64 + col] = acc[q];
      }
    }
  }
  __syncthreads();

  // GEMM2: O(32x32) = Tm(32x64) * Xw(64x32); fuse valid-mask + round + store
  for (int mt = 0; mt < 2; ++mt) {
    for (int nt = 0; nt < 2; ++nt) {
      v8f acc = {};
      for (int ks = 0; ks < 16; ++ks) {
        v2f a  = loadA(Tm, 64, mt * 16, ks * 4);
        v2f bb = loadB(Xw, 32, ks * 4, nt * 16);
        acc = __builtin_amdgcn_wmma_f32_16x16x4_f32(false, a, false, bb,
                                                    (short)0, acc, false, false);
      }
      for (int q = 0; q < 8; ++q) {
        int row = mt * 16 + q + (lh << 3);
        int col = nt * 16 + lr;
        if (row < MASK_H && col < MASK_W) {
          float v = (vy[row] && vx[col]) ? rintf(acc[q]) : 0.0f;  // RNE == jnp.round
          out[row * MASK_W + col] = v;
        }
      }
    }
  }
}

extern "C" void kernel_launch(void* const* d_in, const int* in_sizes, int n_in,
                              void* d_out, int out_size, void* d_ws, size_t ws_size,
                              hipStream_t stream) {
  const float* proposals = (const float*)d_in[0];
  const int*   gt_cls    = (const int*)d_in[1];
  const float* gt_boxes  = (const float*)d_in[2];
  const float* gt_masks  = (const float*)d_in[3];
  const int B = in_sizes[0] / (N_PROP * 4);

  float* f = (float*)d_ws;
  float* pos_score = f;
  float* neg_score = pos_score + (size_t)B * SCORE_PITCH;
  float* roibox    = neg_score + (size_t)B * SCORE_PITCH;
  int*   iw        = (int*)(roibox + (size_t)B * POS_CAP * 4);
  int* pos_idx = iw;
  int* neg_idx = pos_idx + B * POS_CAP;
  int* assignw = neg_idx + B * NEG_CAP;
  int* pn      = assignw + B * POS_CAP;

  dim3 gA((N_PROP + 255) / 256, B);
  k_score<<<gA, 256, 0, stream>>>(proposals, gt_cls, gt_boxes, pos_score, neg_score);
  k_select<<<B, 256, 0, stream>>>(pos_score, neg_score, pos_idx, neg_idx, pn);
  k_pack<<<B, 256, 0, stream>>>(proposals, gt_cls, gt_boxes, pos_idx, neg_idx, pn,
                                assignw, roibox, (float*)d_out, B);
  float* masks_out = (float*)d_out + (size_t)B * TRAIN_ROIS * 9;  // 4+1+4 per slot
  k_masks<<<B * TRAIN_ROIS, 32, 0, stream>>>(gt_masks, pn, assignw, roibox, masks_out);
}